// EdgePredictionGNNModel_75849122447429
// MI455X (gfx1250) — compile-verified
//
#include <hip/hip_runtime.h>

typedef __attribute__((ext_vector_type(2))) float v2f;
typedef __attribute__((ext_vector_type(8))) float v8f;

// Problem constants (match reference)
#define T0C 135168
#define T1C 12288
#define F0C 25
#define F1C 10
#define DC  100
#define HC  256
#define CC  128
#define ICC 4096

// LDS weight tile: 32 K-pairs x 64 cols x 2 (k, k+1), padded pair stride
#define PAIR_STRIDE 130   // dwords per K-pair row (64 cols * 2 + 2 pad)

// ---------------------------------------------------------------------------
// Gather + mean: one wave32 per output row, lane owns columns lane+32*j.
// ---------------------------------------------------------------------------
template<int DIM>
__global__ __launch_bounds__(256)
void gather_mean_kernel(const float* __restrict__ table,
                        const int* __restrict__ ind,
                        float* __restrict__ out,
                        int rows, int fan, float inv_fan) {
    const int wave = threadIdx.x >> 5;
    const int lane = threadIdx.x & 31;
    const int row  = blockIdx.x * (blockDim.x >> 5) + wave;
    if (row >= rows) return;

    constexpr int NCH = (DIM + 31) / 32;
    float acc[NCH];
#pragma unroll
    for (int j = 0; j < NCH; ++j) acc[j] = 0.0f;

    const int* ip = ind + (size_t)row * fan;
    for (int f = 0; f < fan; ++f) {
        const int idx = ip[f];
        if (f + 1 < fan) {
            // hint the next random row into cache (global_prefetch_b8)
            __builtin_prefetch(table + (size_t)ip[f + 1] * DIM, 0, 3);
        }
        const float* rp = table + (size_t)idx * DIM;
#pragma unroll
        for (int j = 0; j < NCH; ++j) {
            const int d = lane + j * 32;
            if (d < DIM) acc[j] += rp[d];
        }
    }
    float* op = out + (size_t)row * DIM;
#pragma unroll
    for (int j = 0; j < NCH; ++j) {
        const int d = lane + j * 32;
        if (d < DIM) op[d] = acc[j] * inv_fan;
    }
}

// ---------------------------------------------------------------------------
// Dual-operand fp32 WMMA GEMM:  Out = act(Aa@Wa + Ab@Wb + bias)
//   - grid.x : M/128 (8 waves per block, one 16-row M-tile per wave)
//   - grid.y : N/64  (each wave computes 16x64 via 4 accumulators)
//   - weights staged to LDS with GLOBAL_LOAD_ASYNC_TO_LDS_B32 (ASYNCcnt),
//     in K-pair interleaved layout so B fragments are single ds_load_b64
//   - V_WMMA_F32_16X16X4_F32, K stepped by 4; EXEC all-ones throughout
// ---------------------------------------------------------------------------
__global__ __launch_bounds__(256)
void gemm_dual_wmma_kernel(const float* __restrict__ Aa, int lda_a,
                           const float* __restrict__ Ab, int lda_b,
                           const float* __restrict__ Wa,
                           const float* __restrict__ Wb,
                           const float* __restrict__ bias,
                           float* __restrict__ Out,
                           int K, int N, int do_relu) {
    __shared__ float lds[32 * PAIR_STRIDE];   // 16.6 KB

    const int lane  = threadIdx.x & 31;
    const int wave  = threadIdx.x >> 5;
    const int half  = lane >> 4;     // 0: lanes 0-15, 1: lanes 16-31
    const int l15   = lane & 15;
    const int nb    = blockIdx.y * 64;
    const int mtile = blockIdx.x * 8 + wave;

    // LDS byte offset of the staging buffer (low 32 bits of generic address)
    const unsigned lds_base = (unsigned)(uintptr_t)(&lds[0]);

    const v8f vzero = {0.f, 0.f, 0.f, 0.f, 0.f, 0.f, 0.f, 0.f};
    v8f acc[4];
#pragma unroll
    for (int t = 0; t < 4; ++t) acc[t] = vzero;

    const int mrow = mtile * 16 + l15;   // A-matrix row this lane feeds

    for (int part = 0; part < 2; ++part) {
        const float* Ap = part ? Ab : Aa;          // uniform across wave
        const float* Wp = part ? Wb : Wa;
        if (Ap == nullptr) continue;               // uniform branch
        const int ldA = part ? lda_b : lda_a;

        for (int k0 = 0; k0 < K; k0 += 64) {
            const int ck = (K - k0 < 64) ? (K - k0) : 64;   // multiple of 4

            __syncthreads();   // previous chunk fully consumed

            // Async-stage W[k0..k0+ck, nb..nb+63] into LDS, K-pair interleaved:
            //   lds[(r>>1)*PAIR_STRIDE + c*2 + (r&1)] = W[k0+r][nb+c]
            {
                const unsigned long long wbase =
                    (unsigned long long)(uintptr_t)(Wp + (size_t)k0 * N + nb);
                for (int i = threadIdx.x; i < ck * 64; i += 256) {
                    const int r = i >> 6, c = i & 63;
                    const unsigned ldsoff =
                        lds_base + 4u * (unsigned)((r >> 1) * PAIR_STRIDE + c * 2 + (r & 1));
                    const unsigned voff = 4u * (unsigned)(r * N + c);
                    asm volatile("global_load_async_to_lds_b32 %0, %1, %2"
                                 :: "v"(ldsoff), "v"(voff), "s"(wbase)
                                 : "memory");
                }
                asm volatile("s_wait_asynccnt 0x0" ::: "memory");
            }
            __syncthreads();   // all waves' async fills visible

            // A fragment base: lanes 0-15 hold K=k,k+1; lanes 16-31 K=k+2,k+3
            const float* arow = Ap + (size_t)mrow * ldA + k0 + half * 2;

            // one-step software pipeline on the A fragment
            v2f a_cur;
            a_cur.x = arow[0];
            a_cur.y = arow[1];
            for (int kk = 0; kk < ck; kk += 4) {
                v2f a_nxt = a_cur;
                if (kk + 4 < ck) {
                    a_nxt.x = arow[kk + 4];
                    a_nxt.y = arow[kk + 5];
                }
                // B pair row: row = kk + half*2  ->  pair = kk/2 + half
                const int pb = ((kk >> 1) + half) * PAIR_STRIDE;
#pragma unroll
                for (int t = 0; t < 4; ++t) {
                    const v2f b = *reinterpret_cast<const v2f*>(
                        &lds[pb + (t * 16 + l15) * 2]);   // ds_load_b64
                    acc[t] = __builtin_amdgcn_wmma_f32_16x16x4_f32(
                        false, a_cur, false, b, (short)0, acc[t], false, false);
                }
                a_cur = a_nxt;
            }
        }
    }

    // C/D layout: VGPR v -> row (v + 8*half), col = lane&15 within subtile
    const int rbase = mtile * 16 + half * 8;
#pragma unroll
    for (int t = 0; t < 4; ++t) {
        const int col = nb + t * 16 + l15;
        const float bv = bias ? bias[col] : 0.0f;
#pragma unroll
        for (int v = 0; v < 8; ++v) {
            float x = acc[t][v] + bv;
            if (do_relu) x = fmaxf(x, 0.0f);
            Out[(size_t)(rbase + v) * N + col] = x;
        }
    }
}

// ---------------------------------------------------------------------------
// E[e,c] = out_unique[rev[e mod IC], c] * out_unique[rev[IC + e], c]
// (pos rows e<IC, neg rows e>=IC both resolve to rev[IC+e])
// ---------------------------------------------------------------------------
__global__ __launch_bounds__(256)
void edge_product_kernel(const float* __restrict__ U,
                         const int* __restrict__ rev,
                         float* __restrict__ E) {
    const int i = blockIdx.x * blockDim.x + threadIdx.x;
    const int e = i >> 7;
    const int c = i & 127;
    if (e >= 2 * ICC) return;
    const int es = (e < ICC) ? e : (e - ICC);
    const int s = rev[es];
    const int d = rev[ICC + e];
    E[i] = U[(size_t)s * CC + c] * U[(size_t)d * CC + c];
}

// ---------------------------------------------------------------------------
// scores[e] = dot(Tm[e,:], Wp2) + bp2 : one wave32 per row, shfl reduction
// ---------------------------------------------------------------------------
__global__ __launch_bounds__(256)
void score_kernel(const float* __restrict__ Tm,
                  const float* __restrict__ Wp2,
                  const float* __restrict__ bp2,
                  float* __restrict__ out, int rows) {
    const int wave = threadIdx.x >> 5;
    const int lane = threadIdx.x & 31;
    const int row  = blockIdx.x * (blockDim.x >> 5) + wave;
    if (row >= rows) return;
    const float* rp = Tm + (size_t)row * CC;
    float acc = 0.0f;
#pragma unroll
    for (int j = 0; j < CC / 32; ++j)
        acc += rp[lane + j * 32] * Wp2[lane + j * 32];
#pragma unroll
    for (int off = 16; off > 0; off >>= 1)
        acc += __shfl_xor(acc, off, 32);
    if (lane == 0) out[row] = acc + bp2[0];
}

// ---------------------------------------------------------------------------
extern "C" void kernel_launch(void* const* d_in, const int* in_sizes, int n_in,
                              void* d_out, int out_size, void* d_ws, size_t ws_size,
                              hipStream_t stream) {
    (void)in_sizes; (void)n_in; (void)out_size; (void)ws_size;

    const float* x_feat = (const float*)d_in[0];
    const float* Wn0    = (const float*)d_in[1];
    const float* Ws0    = (const float*)d_in[2];
    const float* b0     = (const float*)d_in[3];
    const float* Wn1    = (const float*)d_in[4];
    const float* Ws1    = (const float*)d_in[5];
    const float* b1     = (const float*)d_in[6];
    const float* Wp1    = (const float*)d_in[7];
    const float* bp1    = (const float*)d_in[8];
    const float* Wp2    = (const float*)d_in[9];
    const float* bp2    = (const float*)d_in[10];
    const int*   col0   = (const int*)d_in[11];
    const int*   col1   = (const int*)d_in[12];
    const int*   rev    = (const int*)d_in[13];

    // Workspace layout (floats), ~220 MB total
    float* ws     = (float*)d_ws;
    float* neigh0 = ws;                                   // T0 x D
    float* h      = neigh0 + (size_t)T0C * DC;            // T0 x H
    float* neigh1 = h      + (size_t)T0C * HC;            // T1 x H
    float* outu   = neigh1 + (size_t)T1C * HC;            // T1 x C
    float* E      = outu   + (size_t)T1C * CC;            // 2*IC x C
    float* Tm     = E      + (size_t)2 * ICC * CC;        // 2*IC x C
    float* scores = (float*)d_out;                        // 2*IC

    // 1) neigh0 = mean(x_feat[col_ind0])        (bandwidth-dominant step)
    gather_mean_kernel<DC><<<T0C / 8, 256, 0, stream>>>(
        x_feat, col0, neigh0, T0C, F0C, 1.0f / F0C);

    // 2) h = relu(neigh0@Wn0 + x_feat[:T0]@Ws0 + b0)
    gemm_dual_wmma_kernel<<<dim3(T0C / 128, HC / 64), 256, 0, stream>>>(
        neigh0, DC, x_feat, DC, Wn0, Ws0, b0, h, DC, HC, 1);

    // 3) neigh1 = mean(h[col_ind1])
    gather_mean_kernel<HC><<<T1C / 8, 256, 0, stream>>>(
        h, col1, neigh1, T1C, F1C, 1.0f / F1C);

    // 4) out_unique = neigh1@Wn1 + h[:T1]@Ws1 + b1
    gemm_dual_wmma_kernel<<<dim3(T1C / 128, CC / 64), 256, 0, stream>>>(
        neigh1, HC, h, HC, Wn1, Ws1, b1, outu, HC, CC, 0);

    // 5) E = h_src * h_dst via reverse_map
    edge_product_kernel<<<(2 * ICC * CC) / 256, 256, 0, stream>>>(outu, rev, E);

    // 6) Tm = relu(E @ Wp1 + bp1)   (single-operand GEMM)
    gemm_dual_wmma_kernel<<<dim3((2 * ICC) / 128, CC / 64), 256, 0, stream>>>(
        E, CC, nullptr, 0, Wp1, nullptr, bp1, Tm, CC, CC, 1);

    // 7) scores = Tm @ Wp2 + bp2
    score_kernel<<<(2 * ICC) / 8, 256, 0, stream>>>(Tm, Wp2, bp2, scores, 2 * ICC);
}